// LSTM_5557687681204
// MI455X (gfx1250) — compile-verified
//
#include <hip/hip_runtime.h>
#include <math.h>

#define Bsz 64
#define Tsz 512
#define Hsz 1024
#define NWG 64   // persistent workgroups for the serial chain (must be co-resident)

typedef __attribute__((ext_vector_type(2))) float v2f;
typedef __attribute__((ext_vector_type(4))) float v4f;
typedef __attribute__((ext_vector_type(8))) float v8f;

__device__ __forceinline__ float sigf(float x) { return 1.0f / (1.0f + __expf(-x)); }

// ---------------------------------------------------------------------------
// One wave, ONE 16-row M tile, 4 gates sharing the A operand.
//   C_g[m][n] = sum_k X[row0+m][k] * U_g[h0+n][k]
// A layout (16x4 f32): lanes 0-15 -> K={0,1}, lanes 16-31 -> K={2,3}, M=lane&15.
// B layout (4x16 f32): symmetric, N = lane&15.
// ---------------------------------------------------------------------------
__device__ __forceinline__ void wave_gemm4(
    const float* __restrict__ X, long row0, int h0,
    const float* __restrict__ U0, const float* __restrict__ U1,
    const float* __restrict__ U2, const float* __restrict__ U3,
    v8f& c0, v8f& c1, v8f& c2, v8f& c3)
{
    const int lane = threadIdx.x & 31;
    const int mr   = lane & 15;
    const int kh   = (lane >> 4) << 1;  // 0 or 2
    const float* xr = X  + (row0 + mr) * (long)Hsz + kh;
    const float* u0 = U0 + (long)(h0 + mr) * Hsz + kh;
    const float* u1 = U1 + (long)(h0 + mr) * Hsz + kh;
    const float* u2 = U2 + (long)(h0 + mr) * Hsz + kh;
    const float* u3 = U3 + (long)(h0 + mr) * Hsz + kh;
#pragma unroll
    for (int e = 0; e < 8; ++e) { c0[e] = 0.f; c1[e] = 0.f; c2[e] = 0.f; c3[e] = 0.f; }
#pragma unroll 4
    for (int k = 0; k < Hsz; k += 4) {
        v2f a  = *(const v2f*)(xr + k);
        v2f b0 = *(const v2f*)(u0 + k);
        v2f b1 = *(const v2f*)(u1 + k);
        v2f b2 = *(const v2f*)(u2 + k);
        v2f b3 = *(const v2f*)(u3 + k);
        c0 = __builtin_amdgcn_wmma_f32_16x16x4_f32(false, a, false, b0, (short)0, c0, false, false);
        c1 = __builtin_amdgcn_wmma_f32_16x16x4_f32(false, a, false, b1, (short)0, c1, false, false);
        c2 = __builtin_amdgcn_wmma_f32_16x16x4_f32(false, a, false, b2, (short)0, c2, false, false);
        c3 = __builtin_amdgcn_wmma_f32_16x16x4_f32(false, a, false, b3, (short)0, c3, false, false);
    }
}

// ---------------------------------------------------------------------------
// One wave, TWO 16-row M tiles x 4 gates: the four B loads are shared by both
// M tiles (6 loads per 8 WMMAs -> 10.7 FLOP/byte from cache instead of 6.4).
// ---------------------------------------------------------------------------
__device__ __forceinline__ void wave_gemm4x2(
    const float* __restrict__ X, long row0, int h0,
    const float* __restrict__ U0, const float* __restrict__ U1,
    const float* __restrict__ U2, const float* __restrict__ U3,
    v8f& c00, v8f& c01, v8f& c02, v8f& c03,
    v8f& c10, v8f& c11, v8f& c12, v8f& c13)
{
    const int lane = threadIdx.x & 31;
    const int mr   = lane & 15;
    const int kh   = (lane >> 4) << 1;  // 0 or 2
    const float* xr0 = X  + (row0 + mr) * (long)Hsz + kh;
    const float* xr1 = xr0 + 16 * Hsz;
    const float* u0  = U0 + (long)(h0 + mr) * Hsz + kh;
    const float* u1  = U1 + (long)(h0 + mr) * Hsz + kh;
    const float* u2  = U2 + (long)(h0 + mr) * Hsz + kh;
    const float* u3  = U3 + (long)(h0 + mr) * Hsz + kh;
#pragma unroll
    for (int e = 0; e < 8; ++e) {
        c00[e] = 0.f; c01[e] = 0.f; c02[e] = 0.f; c03[e] = 0.f;
        c10[e] = 0.f; c11[e] = 0.f; c12[e] = 0.f; c13[e] = 0.f;
    }
#pragma unroll 2
    for (int k = 0; k < Hsz; k += 4) {
        v2f a0 = *(const v2f*)(xr0 + k);
        v2f a1 = *(const v2f*)(xr1 + k);
        v2f b0 = *(const v2f*)(u0 + k);
        v2f b1 = *(const v2f*)(u1 + k);
        v2f b2 = *(const v2f*)(u2 + k);
        v2f b3 = *(const v2f*)(u3 + k);
        c00 = __builtin_amdgcn_wmma_f32_16x16x4_f32(false, a0, false, b0, (short)0, c00, false, false);
        c10 = __builtin_amdgcn_wmma_f32_16x16x4_f32(false, a1, false, b0, (short)0, c10, false, false);
        c01 = __builtin_amdgcn_wmma_f32_16x16x4_f32(false, a0, false, b1, (short)0, c01, false, false);
        c11 = __builtin_amdgcn_wmma_f32_16x16x4_f32(false, a1, false, b1, (short)0, c11, false, false);
        c02 = __builtin_amdgcn_wmma_f32_16x16x4_f32(false, a0, false, b2, (short)0, c02, false, false);
        c12 = __builtin_amdgcn_wmma_f32_16x16x4_f32(false, a1, false, b2, (short)0, c12, false, false);
        c03 = __builtin_amdgcn_wmma_f32_16x16x4_f32(false, a0, false, b3, (short)0, c03, false, false);
        c13 = __builtin_amdgcn_wmma_f32_16x16x4_f32(false, a1, false, b3, (short)0, c13, false, false);
    }
}

// Phase 1: XU0[g][t][h] = x[0,t,:] @ U_g.T   (batch-0 rows only; feeds the chain)
__global__ __launch_bounds__(256) void xu0_gemm(
    const float* __restrict__ x,
    const float* __restrict__ Ui, const float* __restrict__ Uf,
    const float* __restrict__ Uc, const float* __restrict__ Uo,
    float* __restrict__ XU0)
{
    const int wave = threadIdx.x >> 5, lane = threadIdx.x & 31;
    const long row0 = (long)blockIdx.x * 16;            // t-rows (b == 0)
    const int  h0   = (blockIdx.y * 8 + wave) * 16;
    v8f ci, cf, cc, co;
    wave_gemm4(x, row0, h0, Ui, Uf, Uc, Uo, ci, cf, cc, co);
    const int n  = lane & 15;
    const int mb = (lane >> 4) << 3;
    const int h  = h0 + n;
#pragma unroll
    for (int e = 0; e < 8; ++e) {
        int t = (int)row0 + mb + e;
        XU0[((long)0 * Tsz + t) * Hsz + h] = ci[e];
        XU0[((long)1 * Tsz + t) * Hsz + h] = cf[e];
        XU0[((long)2 * Tsz + t) * Hsz + h] = cc[e];
        XU0[((long)3 * Tsz + t) * Hsz + h] = co[e];
    }
}

// Phase 2: serial chain over t. Persistent grid; one release/acquire global
// barrier per step (monotonic counter, plain-load spin). Each step:
//   A[t][g][h] = hrow(t) @ W_g[h,:] + b_g[h]      (matvec spread over all waves)
//   hrow(t+1)[h] = o*tanh(f*x0 + i*g)             (recomputed per-WG into LDS)
__global__ __launch_bounds__(256) void lstm_chain(
    const float* __restrict__ x, const float* __restrict__ h0in,
    const float* __restrict__ Wi, const float* __restrict__ Wf,
    const float* __restrict__ Wc, const float* __restrict__ Wo,
    const float* __restrict__ bi, const float* __restrict__ bf,
    const float* __restrict__ bc, const float* __restrict__ bo,
    const float* __restrict__ XU0, float* __restrict__ A, int* __restrict__ bar)
{
    __shared__ float hrow[Hsz];
    const int tid = threadIdx.x;
    for (int h = tid; h < Hsz; h += 256) hrow[h] = h0in[h];  // h0 row 0
    __syncthreads();

    const int lane = tid & 31;
    const int gw   = blockIdx.x * 8 + (tid >> 5);            // 0..NWG*8-1 (512 waves)
    const float* Wsel[4] = {Wi, Wf, Wc, Wo};
    const float* bsel[4] = {bi, bf, bc, bo};

    for (int t = 0; t < Tsz; ++t) {
        // 4096 dot products / 512 waves = 8 rows per wave
#pragma unroll
        for (int j = 0; j < 8; ++j) {
            int row = gw * 8 + j;
            int g = row >> 10;
            int h = row & (Hsz - 1);
            const float* wr = Wsel[g] + (long)h * Hsz;
            float s = 0.f;
#pragma unroll
            for (int i = 0; i < 8; ++i) {
                int k = i * 128 + lane * 4;
                v4f w  = *(const v4f*)(wr + k);
                v4f hv = *(const v4f*)(&hrow[k]);
                s += w.x * hv.x + w.y * hv.y + w.z * hv.z + w.w * hv.w;
            }
#pragma unroll
            for (int off = 16; off > 0; off >>= 1) s += __shfl_xor(s, off, 32);
            if (lane == 0) A[((long)t * 4 + g) * Hsz + h] = s + bsel[g][h];
        }
        __syncthreads();
        if (tid == 0) {
            __hip_atomic_fetch_add(bar, 1, __ATOMIC_RELEASE, __HIP_MEMORY_SCOPE_AGENT);
            const int target = NWG * (t + 1);
            while (__hip_atomic_load(bar, __ATOMIC_ACQUIRE, __HIP_MEMORY_SCOPE_AGENT) < target)
                __builtin_amdgcn_s_sleep(8);
        }
        __syncthreads();
        // recompute hrow(t+1) locally (cheap, avoids a second global barrier)
        for (int h = tid; h < Hsz; h += 256) {
            float ai = A[((long)t * 4 + 0) * Hsz + h] + XU0[((long)0 * Tsz + t) * Hsz + h];
            float af = A[((long)t * 4 + 1) * Hsz + h] + XU0[((long)1 * Tsz + t) * Hsz + h];
            float ac = A[((long)t * 4 + 2) * Hsz + h] + XU0[((long)2 * Tsz + t) * Hsz + h];
            float ao = A[((long)t * 4 + 3) * Hsz + h] + XU0[((long)3 * Tsz + t) * Hsz + h];
            float iv = sigf(ai), fv = sigf(af), gv = tanhf(ac), ov = sigf(ao);
            float xv = x[(long)t * Hsz + h];   // x[0, t, h]
            float cv = fv * xv + iv * gv;
            hrow[h] = ov * tanhf(cv);
        }
        __syncthreads();
    }
}

// Phase 3: fused big GEMM + gate activations + cell/hidden outputs.
// One wave per 32x16 tile (2 M-tiles x 4 gates), XU never materialized.
__global__ __launch_bounds__(256) void lstm_fused(
    const float* __restrict__ x,
    const float* __restrict__ Ui, const float* __restrict__ Uf,
    const float* __restrict__ Uc, const float* __restrict__ Uo,
    const float* __restrict__ A,
    float* __restrict__ out, float* __restrict__ hlast, float* __restrict__ clast)
{
    const int wave = threadIdx.x >> 5, lane = threadIdx.x & 31;
    const long row0 = (long)blockIdx.x * 32;                 // rows of X = b*T + t
    const int  h0   = (blockIdx.y * 8 + wave) * 16;
    v8f c00, c01, c02, c03, c10, c11, c12, c13;
    wave_gemm4x2(x, row0, h0, Ui, Uf, Uc, Uo, c00, c01, c02, c03, c10, c11, c12, c13);
    const int n  = lane & 15;
    const int mb = (lane >> 4) << 3;
    const int h  = h0 + n;
#pragma unroll
    for (int mi = 0; mi < 2; ++mi) {
        const v8f& ci = mi ? c10 : c00;
        const v8f& cf = mi ? c11 : c01;
        const v8f& cc = mi ? c12 : c02;
        const v8f& co = mi ? c13 : c03;
#pragma unroll
        for (int e = 0; e < 8; ++e) {
            long r = row0 + mi * 16 + mb + e;
            int  t = (int)(r & (Tsz - 1));
            int  b = (int)(r >> 9);                          // T == 512
            const float* Ath = A + (long)t * 4 * Hsz + h;
            float ai = Ath[0 * Hsz] + ci[e];
            float af = Ath[1 * Hsz] + cf[e];
            float ac = Ath[2 * Hsz] + cc[e];
            float ao = Ath[3 * Hsz] + co[e];
            float iv = sigf(ai), fv = sigf(af), gv = tanhf(ac), ov = sigf(ao);
            float xv = x[r * Hsz + h];
            float cv = fv * xv + iv * gv;
            float hv = ov * tanhf(cv);
            out[r * Hsz + h] = hv;
            if (t == Tsz - 1) {
                hlast[(long)b * Hsz + h] = hv;
                clast[(long)b * Hsz + h] = cv;
            }
        }
    }
}

extern "C" void kernel_launch(void* const* d_in, const int* in_sizes, int n_in,
                              void* d_out, int out_size, void* d_ws, size_t ws_size,
                              hipStream_t stream) {
    const float* x  = (const float*)d_in[0];
    const float* h0 = (const float*)d_in[1];
    // d_in[2] = c0 : unused by the reference math (c_prev is ignored)
    const float* Wi = (const float*)d_in[3];
    const float* Wf = (const float*)d_in[4];
    const float* Wc = (const float*)d_in[5];
    const float* Wo = (const float*)d_in[6];
    const float* bi = (const float*)d_in[7];
    const float* bf = (const float*)d_in[8];
    const float* bc = (const float*)d_in[9];
    const float* bo = (const float*)d_in[10];
    const float* Ui = (const float*)d_in[11];
    const float* Uf = (const float*)d_in[12];
    const float* Uc = (const float*)d_in[13];
    const float* Uo = (const float*)d_in[14];

    char*  ws  = (char*)d_ws;
    int*   bar = (int*)ws;                                             // 256 B (zeroed)
    float* A   = (float*)(ws + 256);                                   // [T][4][H] = 8 MB
    float* XU0 = (float*)(ws + 256 + (size_t)Tsz * 4 * Hsz * sizeof(float)); // [4][T][H] = 8 MB

    float* out   = (float*)d_out;
    float* hlast = out + (size_t)Bsz * Tsz * Hsz;
    float* clast = hlast + (size_t)Bsz * Hsz;

    hipMemsetAsync(bar, 0, 256, stream);
    xu0_gemm<<<dim3(Tsz / 16, Hsz / 16 / 8), 256, 0, stream>>>(x, Ui, Uf, Uc, Uo, XU0);
    lstm_chain<<<NWG, 256, 0, stream>>>(x, h0, Wi, Wf, Wc, Wo, bi, bf, bc, bo, XU0, A, bar);
    lstm_fused<<<dim3(Bsz * Tsz / 32, Hsz / 16 / 8), 256, 0, stream>>>(x, Ui, Uf, Uc, Uo, A,
                                                                       out, hlast, clast);
}